// FMFMNeuron_42262478192730
// MI455X (gfx1250) — compile-verified
//
#include <hip/hip_runtime.h>
#include <stdint.h>

// snntorch Leaky scan: T=50 sequential steps, B=500000 independent neurons.
// HBM-bound streaming kernel (300MB total @ 23.3TB/s ~= 13us floor).
// CDNA5 path: 8-deep async global->LDS prefetch pipeline + NT stores.

typedef float v4f __attribute__((ext_vector_type(4)));
typedef float v2f __attribute__((ext_vector_type(2)));

#define T_STEPS 50
#define NSTAGE  8      // async pipeline depth (power of 2)
#define BLOCK   256    // 8 waves of 32

// Issue one per-lane 16B async copy global->LDS.
// s_wait_dscnt 0 first: the target LDS slot was just read by this wave
// (slot reuse distance == NSTAGE), and async LDS writes are unordered vs
// the wave's own DS reads, so make sure the read completed.
__device__ __forceinline__ void issue_async(uint32_t lds_off, uint32_t gbyte_off,
                                            const float* __restrict__ base) {
  asm volatile("s_wait_dscnt 0\n\t"
               "global_load_async_to_lds_b128 %0, %1, %2"
               :: "v"(lds_off), "v"(gbyte_off), "s"(base)
               : "memory");
}

template <int N>
__device__ __forceinline__ void wait_async() {
  asm volatile("s_wait_asynccnt %0" :: "i"(N) : "memory");
}

__global__ void __launch_bounds__(BLOCK)
snn_leaky_scan_kernel(const float* __restrict__ spikes,  // [T, B, 2]
                      const float* __restrict__ W,       // [2]
                      float* __restrict__ out,           // [T, B]
                      int B) {
  // Per-wave-private staging: stage s, thread tid -> smem[s*BLOCK + tid] (16B).
  __shared__ v4f smem[NSTAGE * BLOCK];

  const int tid = threadIdx.x;
  const int gid = blockIdx.x * BLOCK + tid;
  const uint32_t b = (uint32_t)gid * 2u;   // each thread owns neurons b, b+1
  if ((int)b >= B) return;                  // B is even (500000)

  const float w1 = W[0];
  const float w2 = W[1];
  const uint32_t uB = (uint32_t)B;

  // Low 32 bits of the generic pointer to a __shared__ object == LDS byte offset.
  const uint32_t lds0 = (uint32_t)(uintptr_t)(&smem[tid]);

  // Prologue: fill all NSTAGE stages (t = 0..7). Byte offset of
  // spikes[t][b][0] = 8*(t*B + b); max ~196M < 2^31.
  #pragma unroll
  for (int t = 0; t < NSTAGE; ++t) {
    issue_async(lds0 + (uint32_t)t * (BLOCK * 16u),
                8u * (uint32_t)t * uB + 8u * b, spikes);
  }

  float mem0 = 0.0f, mem1 = 0.0f;

  auto step = [&](int t) {
    // stage t is guaranteed landed by the preceding s_wait_asynccnt
    v4f x = smem[(t & (NSTAGE - 1)) * BLOCK + tid];  // ds_load_b128
    // cur = spikes @ W  (2-element dot, exact fp32)
    float cur0 = x.x * w1 + x.y * w2;
    float cur1 = x.z * w1 + x.w * w2;
    // reset-by-subtraction from previous membrane, then fire
    float r0 = (mem0 > 1.0f) ? 1.0f : 0.0f;
    float r1 = (mem1 > 1.0f) ? 1.0f : 0.0f;
    mem0 = 0.95f * mem0 + cur0 - r0;
    mem1 = 0.95f * mem1 + cur1 - r1;
    v2f spk;
    spk.x = (mem0 > 1.0f) ? 1.0f : 0.0f;
    spk.y = (mem1 > 1.0f) ? 1.0f : 0.0f;
    // write-once output: non-temporal 8B store
    __builtin_nontemporal_store(spk, (v2f*)(out + (size_t)t * uB + b));
  };

  // Main loop: consume stage t, refill slot with stage t+NSTAGE.
  for (int t = 0; t < T_STEPS - NSTAGE; ++t) {
    wait_async<NSTAGE - 1>();   // 8 in flight -> oldest (stage t) complete
    step(t);
    issue_async(lds0 + (uint32_t)((t + NSTAGE) & (NSTAGE - 1)) * (BLOCK * 16u),
                8u * (uint32_t)(t + NSTAGE) * uB + 8u * b, spikes);
  }

  // Epilogue: drain remaining stages with decreasing wait thresholds.
  wait_async<7>(); step(T_STEPS - 8);
  wait_async<6>(); step(T_STEPS - 7);
  wait_async<5>(); step(T_STEPS - 6);
  wait_async<4>(); step(T_STEPS - 5);
  wait_async<3>(); step(T_STEPS - 4);
  wait_async<2>(); step(T_STEPS - 3);
  wait_async<1>(); step(T_STEPS - 2);
  wait_async<0>(); step(T_STEPS - 1);
}

extern "C" void kernel_launch(void* const* d_in, const int* in_sizes, int n_in,
                              void* d_out, int out_size, void* d_ws, size_t ws_size,
                              hipStream_t stream) {
  const float* spikes = (const float*)d_in[0];  // [T, B, 2] f32
  const float* W      = (const float*)d_in[1];  // [2] f32
  float* out          = (float*)d_out;          // [T, B] f32

  const int B = in_sizes[0] / (2 * T_STEPS);
  const int nthreads = B / 2;                   // 2 neurons per thread (B even)
  const int blocks = (nthreads + BLOCK - 1) / BLOCK;

  hipLaunchKernelGGL(snn_leaky_scan_kernel, dim3(blocks), dim3(BLOCK), 0, stream,
                     spikes, W, out, B);
}